// HashEmbeddingEncoder_20306605375914
// MI455X (gfx1250) — compile-verified
//
#include <hip/hip_runtime.h>
#include <cstdint>
#include <cstddef>

// ---------------------------------------------------------------------------
// Multiresolution hash-grid encoder (instant-NGP style) for gfx1250.
//
// Per point, per level: 8 hashed float2 gathers + trilinear blend -> 2 floats.
// Tables (64 MB total) are L2-resident on MI455X (192 MB L2); kernel is
// L2-gather bound. Strategy: wave32, one thread per point, 16 levels fully
// unrolled (8 independent b64 gathers per level in flight), 32-bit hash math
// in pre-shifted byte-offset space (valid because 2^19 | 2^32), async-to-LDS
// staging of the coordinate stream (one shared GVS address + IOFFSET 0/1K/2K),
// and non-temporal b128 output stores so the 128 MB write stream does not
// evict the tables from L2.
// ---------------------------------------------------------------------------

typedef float v2f __attribute__((ext_vector_type(2)));
typedef float v4f __attribute__((ext_vector_type(4)));

#define NLVL    16
#define TBLN    (1u << 19)
#define BLK     256
// Byte-offset-space mask: (TBLN-1) << 3
#define BMASK   (((TBLN) - 1u) << 3)

__global__ __launch_bounds__(BLK) void hashenc_kernel(
    const float* __restrict__ xin,      // [N,3] raw coords in [-1,1]
    const float* __restrict__ tables,   // [16, 2^19, 2] f32
    float* __restrict__ out,            // [N, 32] f32
    int npts) {
  __shared__ float sx[BLK * 3];

  const uint32_t tid   = threadIdx.x;
  const uint32_t blk0  = (uint32_t)blockIdx.x * BLK;   // first point of block
  const uint32_t nflt  = (uint32_t)npts * 3u;          // total floats in xin
  const float*   gsrc  = xin + (size_t)blk0 * 3;

  // ---- Stage this block's 3*256 coordinate floats into LDS (coalesced). ----
  // Fast path (whole block in range — always true when npts % 256 == 0):
  // three async b32 loads sharing one saddr+voffset address, differing only
  // by the instruction immediate offset, which the HW adds to BOTH the global
  // and the LDS address.
  if (blk0 * 3u + 3u * BLK <= nflt) {
#if defined(__gfx1250__) && __has_builtin(__builtin_amdgcn_global_load_async_to_lds_b32)
    __builtin_amdgcn_global_load_async_to_lds_b32(
        (__attribute__((address_space(1))) int*)(gsrc + tid),
        (__attribute__((address_space(3))) int*)&sx[tid],
        /*offset=*/0, /*cpol=*/0);
    __builtin_amdgcn_global_load_async_to_lds_b32(
        (__attribute__((address_space(1))) int*)(gsrc + tid),
        (__attribute__((address_space(3))) int*)&sx[tid],
        /*offset=*/BLK * 4, /*cpol=*/0);
    __builtin_amdgcn_global_load_async_to_lds_b32(
        (__attribute__((address_space(1))) int*)(gsrc + tid),
        (__attribute__((address_space(3))) int*)&sx[tid],
        /*offset=*/2 * BLK * 4, /*cpol=*/0);
    asm volatile("s_wait_asynccnt 0" ::: "memory");
#else
    #pragma unroll
    for (uint32_t k = 0; k < 3; ++k) {
      const uint32_t gi = k * BLK + tid;
      sx[gi] = gsrc[gi];
    }
#endif
  } else {
    // Tail block: plain guarded loads (cold path).
    #pragma unroll
    for (uint32_t k = 0; k < 3; ++k) {
      const uint32_t gi = k * BLK + tid;
      if (blk0 * 3u + gi < nflt) sx[gi] = gsrc[gi];
    }
  }
  __syncthreads();

  const uint32_t gid = blk0 + tid;
  if (gid >= (uint32_t)npts) return;

  // Normalize to [0, 1-1e-6] exactly as the reference (all fp32).
  const float px = fminf(fmaxf((sx[tid * 3 + 0] + 1.0f) * 0.5f, 0.0f), 1.0f - 1e-6f);
  const float py = fminf(fmaxf((sx[tid * 3 + 1] + 1.0f) * 0.5f, 0.0f), 1.0f - 1e-6f);
  const float pz = fminf(fmaxf((sx[tid * 3 + 2] + 1.0f) * 0.5f, 0.0f), 1.0f - 1e-6f);

  // Per-axis hash primes, pre-scaled by 8 (sizeof(float2)) so all corner
  // index math happens directly in byte-offset space:
  //   ((h ^ L) & M) * 8  ==  ((h*8 + c*8) ^ (L*8)) & (M*8)
  // u32 wraparound is harmless because 2^19 | 2^32.
  const uint32_t P0 = 1546061u, P1 = 1005013u, P2c = 1673733u;
  const uint32_t P0b = P0 * 8u, P1b = P1 * 8u, P2b = P2c * 8u;

  float o[NLVL * 2];

  #pragma unroll
  for (int lvl = 0; lvl < NLVL; ++lvl) {
    const float res = (float)(16 << lvl);
    const float sxf = px * res, syf = py * res, szf = pz * res;
    const float flx = floorf(sxf), fly = floorf(syf), flz = floorf(szf);
    const float fx = sxf - flx, fy = syf - fly, fz = szf - flz;
    const uint32_t ix = (uint32_t)flx, iy = (uint32_t)fly, iz = (uint32_t)flz;

    // Base hash, already in byte-offset space (wraps mod 2^32: OK).
    const uint32_t b8 = ix * P0b + iy * P1b + iz * P2b;
    const uint32_t L8 = (uint32_t)lvl << 3;            // (lvl ^ .) pre-shifted

    // Per-level table base is scalar (saddr); corner offsets are 32-bit.
    const char* __restrict__ tb =
        (const char*)tables + (size_t)lvl * (TBLN * 8u);

    #define CORNER(c8) (*(const v2f*)(tb + ((((b8) + (c8)) ^ L8) & BMASK)))
    const v2f e000 = CORNER(0u);
    const v2f e001 = CORNER(P2b);
    const v2f e010 = CORNER(P1b);
    const v2f e011 = CORNER(P1b + P2b);
    const v2f e100 = CORNER(P0b);
    const v2f e101 = CORNER(P0b + P2b);
    const v2f e110 = CORNER(P0b + P1b);
    const v2f e111 = CORNER(P0b + P1b + P2b);
    #undef CORNER

    // Trilinear weights.
    const float gx = 1.0f - fx, gy = 1.0f - fy, gz = 1.0f - fz;
    const float w00 = gx * gy, w01 = gx * fy, w10 = fx * gy, w11 = fx * fy;
    const float w000 = w00 * gz, w001 = w00 * fz;
    const float w010 = w01 * gz, w011 = w01 * fz;
    const float w100 = w10 * gz, w101 = w10 * fz;
    const float w110 = w11 * gz, w111 = w11 * fz;

    o[2 * lvl + 0] = e000.x * w000 + e001.x * w001 + e010.x * w010 + e011.x * w011
                   + e100.x * w100 + e101.x * w101 + e110.x * w110 + e111.x * w111;
    o[2 * lvl + 1] = e000.y * w000 + e001.y * w001 + e010.y * w010 + e011.y * w011
                   + e100.y * w100 + e101.y * w101 + e110.y * w110 + e111.y * w111;
  }

  // Non-temporal b128 stores: keep the 128 MB output stream out of L2 so the
  // 64 MB of hash tables stay resident.
  v4f* __restrict__ op = (v4f*)(out + (size_t)gid * (NLVL * 2));
  #pragma unroll
  for (int q = 0; q < 8; ++q) {
    v4f v = { o[4 * q + 0], o[4 * q + 1], o[4 * q + 2], o[4 * q + 3] };
    __builtin_nontemporal_store(v, op + q);
  }
}

extern "C" void kernel_launch(void* const* d_in, const int* in_sizes, int n_in,
                              void* d_out, int out_size, void* d_ws, size_t ws_size,
                              hipStream_t stream) {
  const float* x      = (const float*)d_in[0];   // [N,3] f32
  const float* tables = (const float*)d_in[1];   // [16, 2^19, 2] f32
  float*       out    = (float*)d_out;           // [N,32] f32

  const int npts   = in_sizes[0] / 3;            // 1048576
  const int blocks = (npts + BLK - 1) / BLK;

  hipLaunchKernelGGL(hashenc_kernel, dim3(blocks), dim3(BLK), 0, stream,
                     x, tables, out, npts);
}